// QuantumKernel_65481071396156
// MI455X (gfx1250) — compile-verified
//
#include <hip/hip_runtime.h>

typedef __attribute__((ext_vector_type(2))) float v2f;
typedef __attribute__((ext_vector_type(8))) float v8f;

#define M_ROWS 8192
#define N_COLS 8192
#define K_DIM  64

// ---------------------------------------------------------------------------
// Kernel 1: row squared-norms. blockDim = (32, 8). Each wave handles one row:
// lane L loads elements L and L+32 (fully coalesced 128B per wave read),
// squares, reduces across the wave32 with xor-shuffles, lane 0 writes.
// ---------------------------------------------------------------------------
__global__ __launch_bounds__(256) void rbf_norms_kernel(
    const float* __restrict__ src, float* __restrict__ dst, int nrows) {
  int row = blockIdx.x * blockDim.y + threadIdx.y;
  if (row >= nrows) return;
  int lane = threadIdx.x;  // 0..31, wave32
  const float* p = src + (size_t)row * K_DIM;
  float v0 = p[lane];
  float v1 = p[lane + 32];
  float s = v0 * v0 + v1 * v1;
#pragma unroll
  for (int off = 16; off > 0; off >>= 1) s += __shfl_xor(s, off, 32);
  if (lane == 0) dst[row] = s;
}

// ---------------------------------------------------------------------------
// Kernel 2: tiled X*Y^T via V_WMMA_F32_16X16X4_F32 + RBF epilogue.
// One wave computes a 16x64 output strip (1 M-tile x 4 N-tiles).
// A-matrix (16x4 f32) layout: lane r=L%16 -> row M=r; lanes 0-15 hold K={0,1},
// lanes 16-31 hold K={2,3} in 2 VGPRs -> per lane load float2 at
// X[(row0+r)*64 + 4*kc + 2*hi]. B = Y^T, so B[k][n] = Y[n][k]: Y rows load
// identically. C/D (16x16 f32, 8 VGPRs): element v is (M = v + 8*hi, N = r).
// ---------------------------------------------------------------------------
__global__ __launch_bounds__(256) void rbf_wmma_kernel(
    const float* __restrict__ X, const float* __restrict__ Y,
    const float* __restrict__ x2, const float* __restrict__ y2,
    float* __restrict__ out) {
  const int wave = threadIdx.x >> 5;
  const int lane = threadIdx.x & 31;
  const int job  = blockIdx.x * 8 + wave;           // 65536 jobs total
  const int tileM = job >> 7;                       // 512 M-tiles
  const int ngrp  = job & 127;                      // 128 N-groups of 64 cols
  const int row0 = tileM * 16;
  const int col0 = ngrp * 64;
  const int hi = lane >> 4;                         // which half-wave
  const int r  = lane & 15;                         // row/col within tile

  // Preload the full K=64 A-strip for this wave's 16 rows: 16 chunks of v2f.
  const v2f* ax = (const v2f*)(X + (size_t)(row0 + r) * K_DIM);
  v2f a[16];
#pragma unroll
  for (int kc = 0; kc < 16; ++kc) a[kc] = ax[2 * kc + hi];

  const v2f* by0 = (const v2f*)(Y + (size_t)(col0 +  0 + r) * K_DIM);
  const v2f* by1 = (const v2f*)(Y + (size_t)(col0 + 16 + r) * K_DIM);
  const v2f* by2 = (const v2f*)(Y + (size_t)(col0 + 32 + r) * K_DIM);
  const v2f* by3 = (const v2f*)(Y + (size_t)(col0 + 48 + r) * K_DIM);

  v8f acc0 = {}, acc1 = {}, acc2 = {}, acc3 = {};

#pragma unroll
  for (int kc = 0; kc < 16; ++kc) {
    const int i = 2 * kc + hi;
    v2f b0 = by0[i];
    v2f b1 = by1[i];
    v2f b2 = by2[i];
    v2f b3 = by3[i];
    // 8 args: (neg_a, A, neg_b, B, c_mod, C, reuse_a, reuse_b)
    acc0 = __builtin_amdgcn_wmma_f32_16x16x4_f32(false, a[kc], false, b0,
                                                 (short)0, acc0, false, false);
    acc1 = __builtin_amdgcn_wmma_f32_16x16x4_f32(false, a[kc], false, b1,
                                                 (short)0, acc1, false, false);
    acc2 = __builtin_amdgcn_wmma_f32_16x16x4_f32(false, a[kc], false, b2,
                                                 (short)0, acc2, false, false);
    acc3 = __builtin_amdgcn_wmma_f32_16x16x4_f32(false, a[kc], false, b3,
                                                 (short)0, acc3, false, false);
  }

  // Epilogue: exp(-max(x2 + y2 - 2*xy, 0)), GAMMA = 1.
  const float yn0 = y2[col0 +  0 + r];
  const float yn1 = y2[col0 + 16 + r];
  const float yn2 = y2[col0 + 32 + r];
  const float yn3 = y2[col0 + 48 + r];

#pragma unroll
  for (int v = 0; v < 8; ++v) {
    const int mrow = row0 + v + 8 * hi;
    const float xn = x2[mrow];
    float* orow = out + (size_t)mrow * N_COLS + col0 + r;

    float s0 = fmaxf(xn + yn0 - 2.0f * acc0[v], 0.0f);
    float s1 = fmaxf(xn + yn1 - 2.0f * acc1[v], 0.0f);
    float s2 = fmaxf(xn + yn2 - 2.0f * acc2[v], 0.0f);
    float s3 = fmaxf(xn + yn3 - 2.0f * acc3[v], 0.0f);
    orow[ 0] = __expf(-s0);
    orow[16] = __expf(-s1);
    orow[32] = __expf(-s2);
    orow[48] = __expf(-s3);
  }
}

extern "C" void kernel_launch(void* const* d_in, const int* in_sizes, int n_in,
                              void* d_out, int out_size, void* d_ws, size_t ws_size,
                              hipStream_t stream) {
  const float* x = (const float*)d_in[0];
  const float* y = (const float*)d_in[1];
  float* out = (float*)d_out;

  // Workspace: 8192 floats of ||x||^2 then 8192 floats of ||y||^2 (64 KB).
  float* x2 = (float*)d_ws;
  float* y2 = x2 + M_ROWS;

  dim3 nb(32, 8);  // one wave per row, 8 rows per block
  rbf_norms_kernel<<<M_ROWS / 8, nb, 0, stream>>>(x, x2, M_ROWS);
  rbf_norms_kernel<<<N_COLS / 8, nb, 0, stream>>>(y, y2, N_COLS);

  // 512 M-tiles * 128 N-groups = 65536 wave-jobs, 8 waves per 256-thread block.
  rbf_wmma_kernel<<<(512 * 128) / 8, 256, 0, stream>>>(x, y, x2, y2, out);
}